// LearnedBlockMask_41626823032999
// MI455X (gfx1250) — compile-verified
//
#include <hip/hip_runtime.h>
#include <cstdint>

// Problem constants (importance: [512, 256, 256] fp32)
#define B_BATCH   512
#define HW        65536
#define K_TOP     49152              // int(0.75 * 65536)
#define NTHREADS  1024
#define PER_THREAD (HW / NTHREADS)       // 64 contiguous elements per thread
#define PER_VEC    (PER_THREAD / 4)      // 16 uint4 per thread
#define VEC_ITERS  (HW / (NTHREADS * 4)) // 16 x b128 per thread, coalesced

typedef int v4i __attribute__((vector_size(16)));
typedef __attribute__((address_space(1))) v4i* as1_v4i;
typedef __attribute__((address_space(3))) v4i* as3_v4i;

__device__ __forceinline__ void wait_async0() {
#if __has_builtin(__builtin_amdgcn_s_wait_asynccnt)
  __builtin_amdgcn_s_wait_asynccnt(0);
#else
  asm volatile("s_wait_asynccnt 0" ::: "memory");
#endif
}

__device__ __forceinline__ unsigned int flip_key(unsigned int u) {
  return (u & 0x80000000u) ? ~u : (u | 0x80000000u);
}

extern "C" __global__ void __launch_bounds__(NTHREADS)
lbm_topk_mask_kernel(const float* __restrict__ imp, float* __restrict__ out)
{
  // Full 256KB row lives in LDS (CDNA5 WGP has 320KB LDS).
  extern __shared__ unsigned int keys[];          // HW entries (256 KB)
  __shared__ unsigned int hist[256];
  __shared__ unsigned int sufx[256];
  __shared__ unsigned int cnt[NTHREADS];
  __shared__ unsigned int sh_prefix;
  __shared__ unsigned int sh_rem;

  const int t = threadIdx.x;
  const int b = blockIdx.x;
  const float* row  = imp + (size_t)b * HW;
  float*       orow = out + (size_t)b * HW;

  // Scalar output: mask.mean() == k/HW exactly (every row has exactly k ones).
  if (b == 0 && t == 0) out[(size_t)B_BATCH * HW] = (float)K_TOP / (float)HW;

  // ---- 1) Async copy row global -> LDS, coalesced 128-bit ----
#if __has_builtin(__builtin_amdgcn_global_load_async_to_lds_b128)
  for (int j = 0; j < VEC_ITERS; ++j) {
    const int idx = (j * NTHREADS + t) * 4;
    __builtin_amdgcn_global_load_async_to_lds_b128(
        (as1_v4i)(uintptr_t)(row + idx),
        (as3_v4i)(unsigned int)(uintptr_t)(keys + idx),
        0, 0);
  }
  wait_async0();
#else
  for (int j = 0; j < VEC_ITERS; ++j) {
    const int idx = (j * NTHREADS + t) * 4;
    *(float4*)((float*)keys + idx) = *(const float4*)(row + idx);
  }
#endif
  if (t == 0) { sh_prefix = 0u; sh_rem = K_TOP; }
  if (t < 256) hist[t] = 0u;
  __syncthreads();

  uint4* keys4 = (uint4*)keys;
  const int base4 = t * PER_VEC;   // contiguous 64-elem chunk => index-ordered ties

  // ---- 2) Key flip fused with the pass-3 (top byte) histogram.
  //      Pass 3 matches every element, so accumulate it during the flip sweep.
  for (int i = 0; i < PER_VEC; ++i) {
    uint4 u = keys4[base4 + i];
    u.x = flip_key(u.x); u.y = flip_key(u.y);
    u.z = flip_key(u.z); u.w = flip_key(u.w);
    keys4[base4 + i] = u;
    atomicAdd(&hist[u.x >> 24], 1u);
    atomicAdd(&hist[u.y >> 24], 1u);
    atomicAdd(&hist[u.z >> 24], 1u);
    atomicAdd(&hist[u.w >> 24], 1u);
  }
  __syncthreads();

  // ---- 3) 4-pass radix select: exact key of the k-th largest ----
  for (int pass = 3; pass >= 0; --pass) {
    const unsigned int curPrefix = sh_prefix;
    const unsigned int rem       = sh_rem;
    const int shift = pass * 8;

    if (pass != 3) {
      // Accumulate histogram of this byte over candidates matching the prefix.
      const unsigned int hiMask = 0xFFFFFFFFu << ((pass + 1) * 8);
      if (t < 256) hist[t] = 0u;
      __syncthreads();
      for (int i = 0; i < PER_VEC; ++i) {
        uint4 k = keys4[base4 + i];
        if ((k.x & hiMask) == curPrefix) atomicAdd(&hist[(k.x >> shift) & 0xFFu], 1u);
        if ((k.y & hiMask) == curPrefix) atomicAdd(&hist[(k.y >> shift) & 0xFFu], 1u);
        if ((k.z & hiMask) == curPrefix) atomicAdd(&hist[(k.z >> shift) & 0xFFu], 1u);
        if ((k.w & hiMask) == curPrefix) atomicAdd(&hist[(k.w >> shift) & 0xFFu], 1u);
      }
      __syncthreads();
    }

    // Parallel inclusive suffix-sum over the 256 bins (Hillis-Steele, 8 steps).
    if (t < 256) sufx[t] = hist[t];
    __syncthreads();
    for (int off = 1; off < 256; off <<= 1) {
      unsigned int v = 0;
      if (t < 256) v = sufx[t] + ((t + off < 256) ? sufx[t + off] : 0u);
      __syncthreads();
      if (t < 256) sufx[t] = v;
      __syncthreads();
    }
    // Unique crossing bin: sum(bins > b) < rem <= sum(bins >= b)
    if (t < 256) {
      const unsigned int inc  = sufx[t];           // sum over bins >= t
      const unsigned int excl = inc - hist[t];     // sum over bins >  t
      if (inc >= rem && excl < rem) {
        sh_prefix = curPrefix | ((unsigned int)t << shift);
        sh_rem    = rem - excl;                    // rank within the ==T group
      }
    }
    __syncthreads();
  }

  const unsigned int T    = sh_prefix;  // key of k-th largest
  const unsigned int need = sh_rem;     // #elements equal to T to include

  // ---- 4) Tie ranking by ascending index (matches top_k tie-break) ----
  unsigned int myc = 0;
  for (int i = 0; i < PER_VEC; ++i) {
    uint4 k = keys4[base4 + i];
    myc += (k.x == T) + (k.y == T) + (k.z == T) + (k.w == T);
  }
  cnt[t] = myc;
  __syncthreads();
  // Parallel inclusive scan over 1024 thread counts (Hillis-Steele, 10 steps).
  for (int off = 1; off < NTHREADS; off <<= 1) {
    unsigned int v = cnt[t] + ((t >= off) ? cnt[t - off] : 0u);
    __syncthreads();
    cnt[t] = v;
    __syncthreads();
  }
  unsigned int rank = cnt[t] - myc;   // exclusive prefix of ==T counts

  // ---- 5) Mask into LDS (reuse keys as float storage, b128) ----
  {
    float4* keysf4 = (float4*)keys;
    for (int i = 0; i < PER_VEC; ++i) {
      uint4 k = keys4[base4 + i];
      float4 m;
      if (k.x > T) m.x = 1.0f; else if (k.x == T) { m.x = (rank < need) ? 1.0f : 0.0f; ++rank; } else m.x = 0.0f;
      if (k.y > T) m.y = 1.0f; else if (k.y == T) { m.y = (rank < need) ? 1.0f : 0.0f; ++rank; } else m.y = 0.0f;
      if (k.z > T) m.z = 1.0f; else if (k.z == T) { m.z = (rank < need) ? 1.0f : 0.0f; ++rank; } else m.z = 0.0f;
      if (k.w > T) m.w = 1.0f; else if (k.w == T) { m.w = (rank < need) ? 1.0f : 0.0f; ++rank; } else m.w = 0.0f;
      keysf4[base4 + i] = m;
    }
  }
  __syncthreads();

  // ---- 6) Async store LDS -> global, coalesced 128-bit ----
#if __has_builtin(__builtin_amdgcn_global_store_async_from_lds_b128)
  for (int j = 0; j < VEC_ITERS; ++j) {
    const int idx = (j * NTHREADS + t) * 4;
    __builtin_amdgcn_global_store_async_from_lds_b128(
        (as1_v4i)(uintptr_t)(orow + idx),
        (as3_v4i)(unsigned int)(uintptr_t)((float*)keys + idx),
        0, 0);
  }
  wait_async0();
#else
  for (int j = 0; j < VEC_ITERS; ++j) {
    const int idx = (j * NTHREADS + t) * 4;
    *(float4*)(orow + idx) = *(float4*)((float*)keys + idx);
  }
#endif
}

extern "C" void kernel_launch(void* const* d_in, const int* in_sizes, int n_in,
                              void* d_out, int out_size, void* d_ws, size_t ws_size,
                              hipStream_t stream) {
  (void)in_sizes; (void)n_in; (void)out_size; (void)d_ws; (void)ws_size;
  const float* imp = (const float*)d_in[0];
  float*       out = (float*)d_out;

  // 256KB dynamic LDS per workgroup (CDNA5 allows up to 320KB/WGP).
  (void)hipFuncSetAttribute((const void*)lbm_topk_mask_kernel,
                            hipFuncAttributeMaxDynamicSharedMemorySize,
                            HW * (int)sizeof(unsigned int));

  lbm_topk_mask_kernel<<<B_BATCH, NTHREADS, HW * sizeof(unsigned int), stream>>>(imp, out);
}